// SimCLRAugmentation_23733989277982
// MI455X (gfx1250) — compile-verified
//
#include <hip/hip_runtime.h>
#include <stdint.h>

typedef float v4f __attribute__((ext_vector_type(4)));

#define TPB 256
#define PPT 4                       // pixels per thread per tile
#define PIX_PER_TILE (TPB * PPT)    // 1024 pixels
#define TILES_PER_BLOCK 4

// ---------------------------------------------------------------------------
// gfx1250 async global->LDS copy (16B per lane), tracked by ASYNCcnt.
// ---------------------------------------------------------------------------
__device__ __forceinline__ void async_load_b128(uint32_t lds_addr, const float* gaddr) {
    asm volatile("global_load_async_to_lds_b128 %0, %1, off"
                 :
                 : "v"(lds_addr), "v"((const void*)gaddr)
                 : "memory");
}
__device__ __forceinline__ uint32_t lds_lo32(const void* p) {
    // flat address of LDS = {shared aperture, lds byte offset}; low 32 bits are
    // the wave-relative LDS address the async VDST operand expects.
    return (uint32_t)(uintptr_t)p;
}

// ---------------------------------------------------------------------------
// Color math (matches the JAX reference semantics), 4 pixels at a time so the
// long hue dependency chain pipelines across independent instances.
// ---------------------------------------------------------------------------
__device__ __forceinline__ float clamp01(float v) { return fminf(fmaxf(v, 0.0f), 1.0f); }
__device__ __forceinline__ float grayof(float r, float g, float b) {
    return fmaf(0.2989f, r, fmaf(0.587f, g, 0.114f * b));
}

__device__ __forceinline__ void op_brightness4(float* r, float* g, float* b, float f) {
#pragma unroll
    for (int j = 0; j < PPT; ++j) {
        r[j] = clamp01(r[j] * f); g[j] = clamp01(g[j] * f); b[j] = clamp01(b[j] * f);
    }
}
__device__ __forceinline__ void op_blend_scalar4(float* r, float* g, float* b, float f, float m) {
    float of = (1.0f - f) * m;
#pragma unroll
    for (int j = 0; j < PPT; ++j) {
        r[j] = clamp01(fmaf(f, r[j], of));
        g[j] = clamp01(fmaf(f, g[j], of));
        b[j] = clamp01(fmaf(f, b[j], of));
    }
}
__device__ __forceinline__ void op_saturation4(float* r, float* g, float* b, float f) {
    float of = 1.0f - f;
#pragma unroll
    for (int j = 0; j < PPT; ++j) {
        float gy = of * grayof(r[j], g[j], b[j]);
        r[j] = clamp01(fmaf(f, r[j], gy));
        g[j] = clamp01(fmaf(f, g[j], gy));
        b[j] = clamp01(fmaf(f, b[j], gy));
    }
}
__device__ __forceinline__ void op_hue4(float* r, float* g, float* b, float f) {
#pragma unroll
    for (int j = 0; j < PPT; ++j) {
        float rr = r[j], gg = g[j], bb = b[j];
        float maxc = fmaxf(rr, fmaxf(gg, bb));
        float minc = fminf(rr, fminf(gg, bb));
        float cr   = maxc - minc;
        bool  eqc  = (cr == 0.0f);
        float crd  = eqc ? 1.0f : cr;
        float inv  = 1.0f / crd;
        float rc = (maxc - rr) * inv;
        float gc = (maxc - gg) * inv;
        float bc = (maxc - bb) * inv;
        float hr = bc - gc;
        float hg = 2.0f + rc - bc;
        float hb = 4.0f + gc - rc;
        float t  = (maxc == rr) ? hr : ((maxc == gg) ? hg : hb);
        float s  = eqc ? 0.0f : (cr / maxc);
        float v  = maxc;
        // mod(mod(t,6)/6 + f, 1) == frac(t/6 + f)
        float h = t * (1.0f / 6.0f) + f;
        h = h - floorf(h);
        float h6 = h * 6.0f;
        float fl = floorf(h6);
        float fr = h6 - fl;
        int   i  = (int)fl;
        i = (i < 0) ? 0 : ((i > 5) ? 5 : i);
        float p  = v * (1.0f - s);
        float q  = v * fmaf(-s, fr, 1.0f);
        float tt = v * fmaf(-s, 1.0f - fr, 1.0f);
        r[j] = (i == 0 || i == 5) ? v : (i == 1 ? q : ((i == 2 || i == 3) ? p : tt));
        g[j] = (i == 1 || i == 2) ? v : (i == 0 ? tt : (i == 3 ? q : p));
        b[j] = (i == 3 || i == 4) ? v : (i == 5 ? q : ((i == 0 || i == 1) ? p : tt));
    }
}

__device__ __forceinline__ void denorm4(const v4f& rv, const v4f& gv, const v4f& bv,
                                        float* r, float* g, float* b) {
#pragma unroll
    for (int j = 0; j < PPT; ++j) {
        r[j] = clamp01(fmaf(rv[j], 0.229f, 0.485f));
        g[j] = clamp01(fmaf(gv[j], 0.224f, 0.456f));
        b[j] = clamp01(fmaf(bv[j], 0.225f, 0.406f));
    }
}

// ---------------------------------------------------------------------------
// Kernel 0: zero the per-image gray-sum accumulators
// ---------------------------------------------------------------------------
__global__ void k_zero(float* p, int n) {
    int i = blockIdx.x * blockDim.x + threadIdx.x;
    if (i < n) p[i] = 0.0f;
}

// ---------------------------------------------------------------------------
// Kernel 1: per-image sum of gray(pre-contrast state). Op dispatch is
// scalar (readfirstlane) and runs once per 4-pixel group. Wave32 shuffle
// reduction + one f32 global atomic per block. x is read with default RT
// hint so it stays resident in the 192MB L2 for pass 2.
// ---------------------------------------------------------------------------
__global__ __launch_bounds__(TPB) void k_gray_mean(
    const float* __restrict__ x, const float* __restrict__ br,
    const float* __restrict__ st, const float* __restrict__ hu,
    const int* __restrict__ order, float* __restrict__ gsum, int HW) {
    int bimg = blockIdx.y;
    const float* img = x + (size_t)bimg * 3 * HW;
    float fb = br[bimg], fs = st[bimg], fh = hu[bimg];
    int o[4];
#pragma unroll
    for (int k = 0; k < 4; ++k) o[k] = __builtin_amdgcn_readfirstlane(order[4 * bimg + k]);

    int   pix   = (blockIdx.x * TPB + threadIdx.x) * PPT;
    float local = 0.0f;
    if (pix < HW) {
        v4f rv = *(const v4f*)(img + 0 * (size_t)HW + pix);
        v4f gv = *(const v4f*)(img + 1 * (size_t)HW + pix);
        v4f bv = *(const v4f*)(img + 2 * (size_t)HW + pix);
        float r[PPT], g[PPT], b[PPT];
        denorm4(rv, gv, bv, r, g, b);
#pragma unroll
        for (int k = 0; k < 4; ++k) {
            int op = o[k];
            if (op == 1) break;                       // contrast: stop here
            if (op == 0) op_brightness4(r, g, b, fb);
            else if (op == 2) op_saturation4(r, g, b, fs);
            else op_hue4(r, g, b, fh);
        }
#pragma unroll
        for (int j = 0; j < PPT; ++j) local += grayof(r[j], g[j], b[j]);
    }
    // wave32 reduction
#pragma unroll
    for (int off = 16; off > 0; off >>= 1) local += __shfl_down(local, off, 32);
    __shared__ float red[TPB / 32];
    int lane = threadIdx.x & 31, wid = threadIdx.x >> 5;
    if (lane == 0) red[wid] = local;
    __syncthreads();
    if (threadIdx.x == 0) {
        float s = 0.0f;
#pragma unroll
        for (int w = 0; w < TPB / 32; ++w) s += red[w];
        atomicAdd(&gsum[bimg], s);
    }
}

// ---------------------------------------------------------------------------
// Kernel 2: full jitter. Double-buffered async global->LDS pipeline: each
// block owns 4 consecutive 1024-pixel tiles; tile t+1 is enqueued with
// global_load_async_to_lds_b128 while tile t is processed; s_wait_asynccnt 3
// guarantees the current tile's 3 channel loads retired (ASYNCcnt retires
// loads in order). Each lane consumes exactly the 16B it enqueued, so only
// the per-wave counter is needed (no barrier). Output uses NT stores to
// protect x's L2 residency.
// ---------------------------------------------------------------------------
__global__ __launch_bounds__(TPB) void k_jitter(
    const float* __restrict__ x, const float* __restrict__ br,
    const float* __restrict__ ct, const float* __restrict__ st,
    const float* __restrict__ hu, const int* __restrict__ order,
    const float* __restrict__ gsum, float* __restrict__ out,
    int HW, int ntiles_img) {
    int bimg = blockIdx.y;
    const float* img  = x + (size_t)bimg * 3 * HW;
    float*       outp = out + (size_t)bimg * 3 * HW;
    float fb = br[bimg], fc = ct[bimg], fs = st[bimg], fh = hu[bimg];
    float m = gsum[bimg] / (float)HW;
    int o[4];
#pragma unroll
    for (int k = 0; k < 4; ++k) o[k] = __builtin_amdgcn_readfirstlane(order[4 * bimg + k]);

    __shared__ float smem[2][3][PIX_PER_TILE];   // 24 KiB double buffer
    int tid   = threadIdx.x;
    int tile0 = blockIdx.x * TILES_PER_BLOCK;

    // prefetch tile 0 into buffer 0
    {
        int pix = tile0 * PIX_PER_TILE + tid * PPT;
        if (tile0 < ntiles_img && pix < HW) {
#pragma unroll
            for (int c = 0; c < 3; ++c)
                async_load_b128(lds_lo32(&smem[0][c][tid * PPT]),
                                img + (size_t)c * HW + pix);
        }
    }

    for (int t = 0; t < TILES_PER_BLOCK; ++t) {
        int tile = tile0 + t;
        if (tile >= ntiles_img) break;
        bool has_next = (t + 1 < TILES_PER_BLOCK) && (tile + 1 < ntiles_img);
        if (has_next) {
            int npix = (tile + 1) * PIX_PER_TILE + tid * PPT;
            if (npix < HW) {
#pragma unroll
                for (int c = 0; c < 3; ++c)
                    async_load_b128(lds_lo32(&smem[(t + 1) & 1][c][tid * PPT]),
                                    img + (size_t)c * HW + npix);
            }
            asm volatile("s_wait_asynccnt 0x3" ::: "memory");  // current tile done
        } else {
            asm volatile("s_wait_asynccnt 0x0" ::: "memory");  // drain
        }

        int pix = tile * PIX_PER_TILE + tid * PPT;
        if (pix < HW) {
            int buf = t & 1;
            v4f rv = *(const v4f*)&smem[buf][0][tid * PPT];
            v4f gv = *(const v4f*)&smem[buf][1][tid * PPT];
            v4f bv = *(const v4f*)&smem[buf][2][tid * PPT];
            float r[PPT], g[PPT], b[PPT];
            denorm4(rv, gv, bv, r, g, b);
#pragma unroll
            for (int k = 0; k < 4; ++k) {
                int opc = o[k];                 // SGPR -> scalar branches
                if (opc == 0) op_brightness4(r, g, b, fb);
                else if (opc == 1) op_blend_scalar4(r, g, b, fc, m);
                else if (opc == 2) op_saturation4(r, g, b, fs);
                else op_hue4(r, g, b, fh);
            }
            v4f ro{}, go{}, bo{};
#pragma unroll
            for (int j = 0; j < PPT; ++j) {
                ro[j] = (r[j] - 0.485f) * (1.0f / 0.229f);
                go[j] = (g[j] - 0.456f) * (1.0f / 0.224f);
                bo[j] = (b[j] - 0.406f) * (1.0f / 0.225f);
            }
            __builtin_nontemporal_store(ro, (v4f*)(outp + 0 * (size_t)HW + pix));
            __builtin_nontemporal_store(go, (v4f*)(outp + 1 * (size_t)HW + pix));
            __builtin_nontemporal_store(bo, (v4f*)(outp + 2 * (size_t)HW + pix));
        }
    }
}

// ---------------------------------------------------------------------------
// Launcher
// ---------------------------------------------------------------------------
extern "C" void kernel_launch(void* const* d_in, const int* in_sizes, int n_in,
                              void* d_out, int out_size, void* d_ws, size_t ws_size,
                              hipStream_t stream) {
    const float* x     = (const float*)d_in[0];
    const float* br    = (const float*)d_in[1];
    const float* ct    = (const float*)d_in[2];
    const float* st    = (const float*)d_in[3];
    const float* hu    = (const float*)d_in[4];
    const int*   order = (const int*)d_in[5];
    float*       out   = (float*)d_out;

    int B  = in_sizes[1];              // 32
    int HW = in_sizes[0] / (3 * B);    // 262144

    float* gsum = (float*)d_ws;        // B floats of scratch

    k_zero<<<(B + 255) / 256, 256, 0, stream>>>(gsum, B);

    int ntiles = (HW + PIX_PER_TILE - 1) / PIX_PER_TILE;              // 256
    k_gray_mean<<<dim3(ntiles, B), TPB, 0, stream>>>(x, br, st, hu, order, gsum, HW);

    int blocks2 = (ntiles + TILES_PER_BLOCK - 1) / TILES_PER_BLOCK;   // 64
    k_jitter<<<dim3(blocks2, B), TPB, 0, stream>>>(x, br, ct, st, hu, order,
                                                   gsum, out, HW, ntiles);
}